// FGLLoss_65103114273337
// MI455X (gfx1250) — compile-verified
//
#include <hip/hip_runtime.h>
#include <hip/hip_bf16.h>
#include <stdint.h>

// ---------------------------------------------------------------------------
// DFL / focal loss streaming reduction for MI455X (gfx1250).
//   pred_dist : (N,4,33) f32   -> 158.4 MB, read once  (memory bound, ~7 us)
//   target_bbox (N,4), ref_points (N,2), iou_weight (N) f32
//   out: scalar mean loss (f32)
// ---------------------------------------------------------------------------

constexpr int   ROW_DW     = 33;            // 33 f32 logits per row
constexpr int   TILE_ROWS  = 128;           // rows staged per tile
constexpr int   TILE_DW    = TILE_ROWS * ROW_DW;   // 4224 dwords = 16896 B
constexpr int   BLOCK      = 128;           // 4 wave32
constexpr int   NBLOCKS    = 1875;          // 9375 tiles / 1875 = 5 tiles each
constexpr float ALPHA      = 0.25f;

// --- CDNA5 async global->LDS copy of one tile (uniform 9 asyncs / thread) ---
// Tile = 4224 dwords, flat: 8 x B128 chunks (1024 chunks, lane-consecutive
// 16B -> 512B per wave instruction) + 1 x B32 tail dword per thread.
__device__ __forceinline__ void async_copy_tile(const float* __restrict__ gbase,
                                                const float* lbuf, int t) {
  // addrspace(3) offsets live in the low 32 bits of the generic pointer
  unsigned lbase = (unsigned)(unsigned long long)lbuf;
#pragma unroll
  for (int j = 0; j < 8; ++j) {
    unsigned off   = (unsigned)((j * BLOCK + t) * 16);
    unsigned laddr = lbase + off;
    asm volatile("global_load_async_to_lds_b128 %0, %1, %2"
                 :: "v"(laddr), "v"(off), "s"(gbase) : "memory");
  }
  {
    unsigned off   = 16384u + 4u * (unsigned)t;   // dwords 4096..4223
    unsigned laddr = lbase + off;
    asm volatile("global_load_async_to_lds_b32 %0, %1, %2"
                 :: "v"(laddr), "v"(off), "s"(gbase) : "memory");
  }
}

__device__ __forceinline__ float xlogy_(float w) {
  return (w > 0.0f) ? w * __logf(w) : 0.0f;
}

__global__ __launch_bounds__(BLOCK) void fgl_main(
    const float* __restrict__ pred,   // (R,33)
    const float* __restrict__ tb,     // (N,4)
    const float* __restrict__ rp,     // (N,2)
    const float* __restrict__ iw,     // (N,)
    float* __restrict__ partial,      // (NBLOCKS,)
    int nTiles) {
  __shared__ __align__(16) float lds[2][TILE_DW];   // double buffer, 33 KB
  __shared__ float red[BLOCK];

  const int t = threadIdx.x;
  float acc = 0.0f;

  int tile = blockIdx.x;
  int buf  = 0;
  if (tile < nTiles)
    async_copy_tile(pred + (size_t)tile * TILE_DW, &lds[0][0], t);

  for (; tile < nTiles; tile += (int)gridDim.x) {
    const int  next    = tile + (int)gridDim.x;
    const bool hasNext = next < nTiles;
    if (hasNext)
      async_copy_tile(pred + (size_t)next * TILE_DW, &lds[buf ^ 1][0], t);

    // Each thread has exactly 9 asyncs in flight per tile; in-order
    // completion => waiting to <=9 retires the *current* tile's copies
    // while the prefetch stays in flight.
    if (hasNext) asm volatile("s_wait_asynccnt 9" ::: "memory");
    else         asm volatile("s_wait_asynccnt 0" ::: "memory");
    __syncthreads();

    // ---- compute one row per thread (33-dword stride: bank-conflict-free)
    const float* row = &lds[buf][t * ROW_DW];
    float x[ROW_DW];
#pragma unroll
    for (int j = 0; j < ROW_DW; ++j) x[j] = row[j];

    float m = x[0];
#pragma unroll
    for (int j = 1; j < ROW_DW; ++j) m = fmaxf(m, x[j]);
    float s = 0.0f;
#pragma unroll
    for (int j = 0; j < ROW_DW; ++j) s += __expf(x[j] - m);
    const float logZ = m + __logf(s);

    const int rowId = tile * TILE_ROWS + t;
    const int n     = rowId >> 2;
    const int side  = rowId & 3;

    const float px = rp[2 * n + 0], py = rp[2 * n + 1];
    const float x1 = tb[4 * n + 0], y1 = tb[4 * n + 1];
    const float x2 = tb[4 * n + 2], y2 = tb[4 * n + 3];
    const float c0 = (side & 2) ? py : px;
    const float lo = (side & 2) ? y1 : x1;
    const float hi = (side & 2) ? y2 : x2;
    float d = (side & 1) ? (hi - c0) : (c0 - lo);
    d = fminf(fmaxf(d, 0.0f), 32.0f);

    const int   di   = (int)d;                   // floor, d >= 0
    const float frac = d - (float)di;
    const int   up   = min(di + 1, 32);
    const float w1   = frac;
    const float w0   = (up == di) ? 0.0f : (1.0f - frac);
    const float wgt  = ((d == 32.0f) ? 0.5f : 1.0f) * iw[n];

    const float lp0 = row[di] - logZ;
    const float lp1 = row[up] - logZ;
    const float ce  = xlogy_(w0) + xlogy_(w1) - w0 * lp0 - w1 * lp1;
    float pt = w0 * __expf(lp0) + w1 * __expf(lp1);
    pt = fminf(fmaxf(pt, 1e-6f), 1.0f);
    const float om = 1.0f - pt;
    acc += ALPHA * om * om * ce * wgt;

    __syncthreads();   // before prefetch overwrites this buffer
    buf ^= 1;
  }

  // deterministic block reduction
  red[t] = acc;
  __syncthreads();
#pragma unroll
  for (int o = BLOCK / 2; o > 0; o >>= 1) {
    if (t < o) red[t] += red[t + o];
    __syncthreads();
  }
  if (t == 0) partial[blockIdx.x] = red[0];
}

// ---------------------------------------------------------------------------
// Finalizer: 1 wave sums the 1875 block partials. Lane reduction 32->16 via
// a ones-matrix V_WMMA_F32_16X16X4_F32 (D[m][n] = acc[n] + acc[n+16]), then
// 4 shfl_xor steps. Exact f32 throughout.
// ---------------------------------------------------------------------------
typedef __attribute__((ext_vector_type(2))) float v2f;
typedef __attribute__((ext_vector_type(8))) float v8f;

__global__ __launch_bounds__(32) void fgl_finalize(
    const float* __restrict__ partial, float* __restrict__ out,
    int nPartials, float invCount) {
  const int lane = threadIdx.x;
  float acc = 0.0f;
  for (int i = lane; i < nPartials; i += 32) acc += partial[i];

  v2f a; a.x = 1.0f; a.y = 1.0f;     // A = ones 16x4
  v2f b; b.x = acc;  b.y = 0.0f;     // B rows K0/K1 carry the 32 lane accs
  v8f c = {};
  c = __builtin_amdgcn_wmma_f32_16x16x4_f32(
      /*neg_a=*/false, a, /*neg_b=*/false, b,
      /*c_mod=*/(short)0, c, /*reuse_a=*/false, /*reuse_b=*/false);
  float sum = c[0];                  // = acc[lane&15] + acc[(lane&15)+16]
  sum += __shfl_xor(sum, 1, 32);
  sum += __shfl_xor(sum, 2, 32);
  sum += __shfl_xor(sum, 4, 32);
  sum += __shfl_xor(sum, 8, 32);
  if (lane == 0) out[0] = sum * invCount;
}

extern "C" void kernel_launch(void* const* d_in, const int* in_sizes, int n_in,
                              void* d_out, int out_size, void* d_ws, size_t ws_size,
                              hipStream_t stream) {
  const float* pred = (const float*)d_in[0];
  const float* tb   = (const float*)d_in[1];
  const float* rp   = (const float*)d_in[2];
  const float* iw   = (const float*)d_in[3];

  const int rows   = in_sizes[3] * 4;        // 1,200,000
  const int nTiles = rows / TILE_ROWS;       // 9375 (exact)
  float* partial = (float*)d_ws;             // NBLOCKS floats

  fgl_main<<<NBLOCKS, BLOCK, 0, stream>>>(pred, tb, rp, iw, partial, nTiles);
  fgl_finalize<<<1, 32, 0, stream>>>(partial, (float*)d_out, NBLOCKS,
                                     1.0f / (float)rows);
}